// YOLOv5_84370337562910
// MI455X (gfx1250) — compile-verified
//
#include <hip/hip_runtime.h>
#include <stdint.h>

#define B_IMG   32
#define N_ANCH  25200
#define NCLS    80
#define PSTRIDE 85
#define KCAND   2048
#define CAP     8192
#define MAXDET  300
#define CONF_F  0.001f
#define IOU_F   0.6f
#define NBINS   4096
#define NTILE   128                 // 2048/16
#define NTILES  (NTILE*(NTILE+1)/2) // 8256 upper-triangle tiles

typedef __attribute__((ext_vector_type(2))) float v2f;
typedef __attribute__((ext_vector_type(8))) float v8f;
typedef unsigned int  u32;
typedef unsigned long long u64;
typedef unsigned short u16;

// ---------------- workspace layout (bytes) ----------------
#define OFF_HIST   0ull
#define SZ_HIST    ((size_t)B_IMG*NBINS*4)
#define OFF_CCOUNT (OFF_HIST+SZ_HIST)
#define SZ_CCOUNT  ((size_t)B_IMG*4)
#define OFF_TBITS  (OFF_CCOUNT+SZ_CCOUNT)
#define SZ_TBITS   ((size_t)B_IMG*4)
#define OFF_VALID  ((OFF_TBITS+SZ_TBITS+255ull)&~255ull)
#define SZ_VALID   ((size_t)B_IMG*64*4)
#define OFF_OVER   ((OFF_VALID+SZ_VALID+255ull)&~255ull)
#define SZ_OVER    ((size_t)B_IMG*KCAND*128*2)      // 16 MB, u16 halfwords
#define OFF_KEYS   (OFF_OVER+SZ_OVER)
#define SZ_KEYS    ((size_t)B_IMG*CAP*8)
#define OFF_CBOX   (OFF_KEYS+SZ_KEYS)
#define SZ_CBOX    ((size_t)B_IMG*KCAND*16)
#define OFF_CSCORE (OFF_CBOX+SZ_CBOX)
#define SZ_CSCORE  ((size_t)B_IMG*KCAND*4)
#define OFF_CAREA  (OFF_CSCORE+SZ_CSCORE)
#define OFF_CCLS   (OFF_CAREA+SZ_CSCORE)
#define ZERO_BYTES (OFF_OVER+SZ_OVER)               // zero hist..over each call

// ---------------- kernels ----------------

__global__ void zero_kernel(u32* p, size_t nwords) {
  size_t i = (size_t)blockIdx.x * blockDim.x + threadIdx.x;
  size_t stride = (size_t)gridDim.x * blockDim.x;
  for (; i < nwords; i += stride) p[i] = 0u;
}

// Pass 1: histogram of per-anchor max score (obj * max cls), bins = float bits[30:19]
__global__ __launch_bounds__(256) void hist_kernel(const float* __restrict__ pred,
                                                   u32* __restrict__ hist) {
  __shared__ u32 lh[NBINS];
  const int tid = threadIdx.x;
  const int img = blockIdx.y;
  for (int i = tid; i < NBINS; i += 256) lh[i] = 0u;
  __syncthreads();
  const int a = blockIdx.x * 256 + tid;
  if (a < N_ANCH) {
    const float* p = pred + ((size_t)img * N_ANCH + a) * PSTRIDE;
    float obj = p[4];
    float m = 0.0f;
    #pragma unroll 8
    for (int c = 0; c < NCLS; ++c) m = fmaxf(m, p[5 + c]);
    float s = obj * m;
    if (s > CONF_F) atomicAdd(&lh[__float_as_uint(s) >> 19], 1u);
  }
  __syncthreads();
  for (int i = tid; i < NBINS; i += 256) {
    u32 v = lh[i];
    if (v) atomicAdd(&hist[(size_t)img * NBINS + i], v);
  }
}

// Pass 2: find threshold bin: largest score bin b s.t. #anchors with max >= low-edge(b) >= KCAND
__global__ __launch_bounds__(256) void thresh_kernel(const u32* __restrict__ hist,
                                                     u32* __restrict__ tbits) {
  __shared__ u32 ps[256];
  const int tid = threadIdx.x;
  const int img = blockIdx.x;
  const u32* h = hist + (size_t)img * NBINS;
  u32 part = 0;
  #pragma unroll
  for (int k = 0; k < 16; ++k) part += h[4095 - (tid * 16 + k)];
  ps[tid] = part;
  __syncthreads();
  for (int off = 1; off < 256; off <<= 1) {
    u32 v = (tid >= off) ? ps[tid - off] : 0u;
    __syncthreads();
    ps[tid] += v;
    __syncthreads();
  }
  u32 incl = ps[tid];
  u32 before = incl - part;
  if (before < KCAND && incl >= KCAND) {
    u32 acc = before;
    for (int k = 0; k < 16; ++k) {
      int b = 4095 - (tid * 16 + k);
      acc += h[b];
      if (acc >= KCAND) { tbits[img] = ((u32)b) << 19; break; }
    }
  }
  // if total < KCAND, tbits stays 0 (pre-zeroed) -> CONF floor applies in compact
}

// Pass 3: compact all (anchor,class) with score > CONF and bits >= threshold into key buffer
__global__ __launch_bounds__(256) void compact_kernel(const float* __restrict__ pred,
                                                      const u32* __restrict__ tbits,
                                                      u32* __restrict__ ccount,
                                                      u64* __restrict__ keys) {
  const int img = blockIdx.y;
  const int a = blockIdx.x * 256 + threadIdx.x;
  if (a >= N_ANCH) return;
  const float* p = pred + ((size_t)img * N_ANCH + a) * PSTRIDE;
  const float obj = p[4];
  const u32 T = tbits[img];
  #pragma unroll 4
  for (int c = 0; c < NCLS; ++c) {
    float s = obj * p[5 + c];
    u32 b = __float_as_uint(s);
    if (s > CONF_F && b >= T) {
      u32 pos = atomicAdd(&ccount[img], 1u);
      if (pos < CAP) {
        u32 flat = (u32)a * NCLS + (u32)c;
        keys[(size_t)img * CAP + pos] = ((u64)b << 32) | (u64)(0xFFFFFFFFu - flat);
      }
    }
  }
}

// Pass 4: per-image bitonic sort (descending) of up to 8192 keys; gather top-2048 candidates
__global__ __launch_bounds__(256) void sort_gather_kernel(const float* __restrict__ pred,
                                                          const u64* __restrict__ keys,
                                                          const u32* __restrict__ ccount,
                                                          float4* __restrict__ cbox,
                                                          float* __restrict__ cscore,
                                                          float* __restrict__ carea,
                                                          int*   __restrict__ ccls,
                                                          u32*   __restrict__ validm) {
  __shared__ u64 sk[CAP];   // 64 KB LDS
  const int tid = threadIdx.x;
  const int img = blockIdx.x;
  u32 cnt = ccount[img];
  if (cnt > CAP) cnt = CAP;
  for (int i = tid; i < CAP; i += 256)
    sk[i] = (i < (int)cnt) ? keys[(size_t)img * CAP + i] : 0ull;
  __syncthreads();
  for (int k = 2; k <= CAP; k <<= 1) {
    for (int j = k >> 1; j > 0; j >>= 1) {
      for (int i = tid; i < CAP; i += 256) {
        int l = i ^ j;
        if (l > i) {
          u64 A = sk[i], Bv = sk[l];
          bool up = ((i & k) == 0);            // descending overall
          bool sw = up ? (A < Bv) : (A > Bv);
          if (sw) { sk[i] = Bv; sk[l] = A; }
        }
      }
      __syncthreads();
    }
  }
  const u32 confBits = __float_as_uint(CONF_F);
  for (int i = tid; i < KCAND; i += 256) {
    u64 key = sk[i];
    u32 b = (u32)(key >> 32);
    bool valid = (b > confBits);
    float x1 = 0, y1 = 0, x2 = 0, y2 = 0, sc = 0, ar = 0;
    int cls = 0;
    if (valid) {
      u32 flat = 0xFFFFFFFFu - (u32)(key & 0xFFFFFFFFull);
      u32 anc = flat / NCLS;
      cls = (int)(flat % NCLS);
      const float* p = pred + ((size_t)img * N_ANCH + anc) * PSTRIDE;
      float x = p[0], y = p[1], w = p[2], h = p[3];
      x1 = x - w * 0.5f; y1 = y - h * 0.5f;
      x2 = x + w * 0.5f; y2 = y + h * 0.5f;
      sc = __uint_as_float(b);
      ar = (x2 - x1) * (y2 - y1);
    }
    size_t ci = (size_t)img * KCAND + i;
    cbox[ci] = make_float4(x1, y1, x2, y2);
    cscore[ci] = sc; carea[ci] = ar; ccls[ci] = cls;
    u32 vm = (u32)__ballot(valid);               // wave32 ballot
    if ((tid & 31) == 0) validm[(size_t)img * 64 + (i >> 5)] = vm;
  }
}

// Pass 5: suppression bit-matrix, one wave per 16x16 tile.
// WMMA computes rank-2 areaA[M]+areaB[N]; VALU computes intersections (branchless).
__global__ __launch_bounds__(256) void over_kernel(const float4* __restrict__ cbox,
                                                   const float* __restrict__ carea,
                                                   const int*   __restrict__ ccls,
                                                   u16* __restrict__ over) {
  const int img  = blockIdx.y;
  const int lane = threadIdx.x & 31;
  const int wv   = threadIdx.x >> 5;
  int t = blockIdx.x * 8 + wv;
  if (t >= NTILES) return;                       // never taken (1032*8 == 8256), uniform per wave
  int ti = 0, rem = t;
  while (rem >= NTILE - ti) { rem -= NTILE - ti; ++ti; }
  int tj = ti + rem;

  const size_t base = (size_t)img * KCAND;
  float aA = 0.0f, aB = 0.0f, one = 0.0f;
  if (lane < 16) {
    aA = carea[base + ti * 16 + lane];
    aB = carea[base + tj * 16 + lane];
    one = 1.0f;
  }
#if __has_builtin(__builtin_amdgcn_wmma_f32_16x16x4_f32)
  v2f A = { aA, one };                           // A[M,0]=areaA, A[M,1]=1 (lanes 0-15); K=2,3 zero
  v2f Bm = { one, aB };                          // B[0,N]=1,     B[1,N]=areaB
  v8f C = { 0.f, 0.f, 0.f, 0.f, 0.f, 0.f, 0.f, 0.f };
  C = __builtin_amdgcn_wmma_f32_16x16x4_f32(false, A, false, Bm, (short)0, C,
                                            false, false);
#else
  v8f C = { 0.f, 0.f, 0.f, 0.f, 0.f, 0.f, 0.f, 0.f };
#endif

  const int jN = tj * 16 + (lane & 15);
  const float4 bj = cbox[base + jN];
  const int clsB = ccls[base + jN];
  const float areaBv = carea[base + jN];
  (void)areaBv;

  #pragma unroll
  for (int r = 0; r < 8; ++r) {
    const int iM = ti * 16 + r + ((lane >> 4) << 3);
    const float4 bi = cbox[base + iM];           // 16-lane broadcast load
    const int clsA = ccls[base + iM];
    float iw = fminf(bi.z, bj.z) - fmaxf(bi.x, bj.x);
    float ih = fminf(bi.w, bj.w) - fmaxf(bi.y, bj.y);
    float inter = fmaxf(iw, 0.0f) * fmaxf(ih, 0.0f);
#if __has_builtin(__builtin_amdgcn_wmma_f32_16x16x4_f32)
    float sum = C[r];                            // areaA[M] + areaB[N] from WMMA
#else
    float sum = carea[base + iM] + areaBv;
#endif
    // branchless: bitwise & avoids exec-mask divergence in the hot loop
    bool sup = (bool)((clsA == clsB) & (inter * (1.0f + IOU_F) > IOU_F * sum) & (jN > iM));
    u32 mask = (u32)__ballot(sup);
    if (lane == 0)
      over[(base + (size_t)ti * 16 + r) * 128 + tj] = (u16)(mask & 0xFFFFu);
    if (lane == 16)
      over[(base + (size_t)ti * 16 + r + 8) * 128 + tj] = (u16)(mask >> 16);
  }
}

// Pass 6: one wave32 per image: serial greedy scan over 2048 rows (keep mask in 2 VGPRs/lane),
// then in-order compaction of the first 300 survivors into d_out.
__global__ __launch_bounds__(32) void nms_scan_kernel(const u32* __restrict__ validm,
                                                      const u32* __restrict__ over32,
                                                      const float4* __restrict__ cbox,
                                                      const float* __restrict__ cscore,
                                                      const int*   __restrict__ ccls,
                                                      float* __restrict__ out) {
  const int lane = threadIdx.x;
  const int img = blockIdx.x;
  u32 k0 = validm[(size_t)img * 64 + lane];
  u32 k1 = validm[(size_t)img * 64 + lane + 32];
  const u32* ov = over32 + (size_t)img * KCAND * 64;  // 64 u32 per row

  // first half: keep bit for i lives in k0
  for (int i = 0; i < KCAND / 2; ++i) {
    const u32* rp = ov + (size_t)i * 64;
    __builtin_prefetch(rp + 16 * 64 + lane, 0, 0);
    u32 r0 = rp[lane];
    u32 r1 = rp[lane + 32];
    u32 word = (u32)__shfl((int)k0, i >> 5, 32);
    u32 bit = (word >> (i & 31)) & 1u;
    u32 m = (u32)0 - bit;
    k0 &= ~(r0 & m);
    k1 &= ~(r1 & m);
  }
  // second half: keep bit for i lives in k1
  for (int i = KCAND / 2; i < KCAND; ++i) {
    const u32* rp = ov + (size_t)i * 64;
    int pf = (i + 16 < KCAND) ? (i + 16) : (KCAND - 1);
    __builtin_prefetch(ov + (size_t)pf * 64 + lane, 0, 0);
    u32 r0 = rp[lane];
    u32 r1 = rp[lane + 32];
    u32 word = (u32)__shfl((int)k1, (i >> 5) - 32, 32);
    u32 bit = (word >> (i & 31)) & 1u;
    u32 m = (u32)0 - bit;
    k0 &= ~(r0 & m);
    k1 &= ~(r1 & m);
  }

  // prefix popcount ranks across the 64 keep words
  int c0 = __popc(k0), c1 = __popc(k1);
  int p0 = c0, p1 = c1;
  #pragma unroll
  for (int off = 1; off < 32; off <<= 1) {
    int v0 = __shfl_up(p0, off, 32);
    int v1 = __shfl_up(p1, off, 32);
    if (lane >= off) { p0 += v0; p1 += v1; }
  }
  int tot0 = __shfl(p0, 31, 32);
  int e0 = p0 - c0;
  int e1 = tot0 + (p1 - c1);

  const size_t base = (size_t)img * KCAND;
  float* ob = out + (size_t)img * MAXDET * 6;
  int r = e0;
  u32 m = k0;
  while (m) {
    int b = __ffs(m) - 1; m &= m - 1;
    if (r < MAXDET) {
      int ci = lane * 32 + b;
      float4 bx = cbox[base + ci];
      float* o = ob + (size_t)r * 6;
      o[0] = bx.x; o[1] = bx.y; o[2] = bx.z; o[3] = bx.w;
      o[4] = cscore[base + ci];
      o[5] = (float)ccls[base + ci];
    }
    ++r;
  }
  r = e1;
  m = k1;
  while (m) {
    int b = __ffs(m) - 1; m &= m - 1;
    if (r < MAXDET) {
      int ci = 1024 + lane * 32 + b;
      float4 bx = cbox[base + ci];
      float* o = ob + (size_t)r * 6;
      o[0] = bx.x; o[1] = bx.y; o[2] = bx.z; o[3] = bx.w;
      o[4] = cscore[base + ci];
      o[5] = (float)ccls[base + ci];
    }
    ++r;
  }
}

// ---------------- launch ----------------
extern "C" void kernel_launch(void* const* d_in, const int* in_sizes, int n_in,
                              void* d_out, int out_size, void* d_ws, size_t ws_size,
                              hipStream_t stream) {
  const float* pred = (const float*)d_in[0];
  float* out = (float*)d_out;
  char* ws = (char*)d_ws;

  u32* hist     = (u32*)(ws + OFF_HIST);
  u32* ccount   = (u32*)(ws + OFF_CCOUNT);
  u32* tbits    = (u32*)(ws + OFF_TBITS);
  u32* validm   = (u32*)(ws + OFF_VALID);
  u16* over     = (u16*)(ws + OFF_OVER);
  u32* over32   = (u32*)(ws + OFF_OVER);
  u64* keys     = (u64*)(ws + OFF_KEYS);
  float4* cbox  = (float4*)(ws + OFF_CBOX);
  float* cscore = (float*)(ws + OFF_CSCORE);
  float* carea  = (float*)(ws + OFF_CAREA);
  int*   ccls   = (int*)(ws + OFF_CCLS);

  const int ablocks = (N_ANCH + 255) / 256;   // 99

  zero_kernel<<<4096, 256, 0, stream>>>((u32*)ws, ZERO_BYTES / 4);
  zero_kernel<<<64, 256, 0, stream>>>((u32*)out, (size_t)B_IMG * MAXDET * 6);

  hist_kernel<<<dim3(ablocks, B_IMG), 256, 0, stream>>>(pred, hist);
  thresh_kernel<<<B_IMG, 256, 0, stream>>>(hist, tbits);
  compact_kernel<<<dim3(ablocks, B_IMG), 256, 0, stream>>>(pred, tbits, ccount, keys);
  sort_gather_kernel<<<B_IMG, 256, 0, stream>>>(pred, keys, ccount,
                                                cbox, cscore, carea, ccls, validm);
  over_kernel<<<dim3(NTILES / 8, B_IMG), 256, 0, stream>>>(cbox, carea, ccls, over);
  nms_scan_kernel<<<B_IMG, 32, 0, stream>>>(validm, over32, cbox, cscore, ccls, out);
}